// PoissonDirichlet_9088150798493
// MI455X (gfx1250) — compile-verified
//
#include <hip/hip_runtime.h>

// Pitman-Yor mixture: fused loglik (WMMA) + softmax + noise-weighted sample.
// Bandwidth-bound on noise [N,100,12] f32 = 629 MB -> single streaming pass.

#define NUM_FEAT 12
#define MAX_K    100
#define KPAD     112          // 7 WMMA column tiles of 16
#define EPSV     1e-10f

typedef float v2f __attribute__((ext_vector_type(2)));
typedef float v8f __attribute__((ext_vector_type(8)));

// workspace layout (floats)
//   coef  : [KPAD][24]  interleaved (-0.5/s2, mu/s2) per feature  (WMMA B)
//   cvec  : [KPAD]      norm_const + log_w - 0.5*sum(mu^2/s2); -1e30 pad
//   mustd : [MAX_K][24] mu[0..11], std[0..11]
#define WS_COEF  0
#define WS_CVEC  (KPAD * 24)
#define WS_MUSTD (WS_CVEC + KPAD)

__global__ void pd_precompute(const float* __restrict__ mu,
                              const float* __restrict__ var,
                              float* __restrict__ ws)
{
    int k = threadIdx.x;
    if (k >= KPAD) return;
    float* coef  = ws + WS_COEF;
    float* cvec  = ws + WS_CVEC;
    float* mustd = ws + WS_MUSTD;
    if (k >= MAX_K) {                       // pad tiles: zero GEMM, -inf logit
        cvec[k] = -1e30f;
        for (int d = 0; d < 24; ++d) coef[k * 24 + d] = 0.0f;
        return;
    }
    // stick-breaking log-weight, a=0.5 b=1.0: v_j = 0.5/(1.5+0.5j)
    float logrem = 0.0f;
    for (int j = 0; j < k; ++j)
        logrem += log1pf(-(0.5f / (1.5f + 0.5f * (float)j)));
    float vk = 0.5f / (1.5f + 0.5f * (float)k);
    float c  = logf(vk * __expf(logrem) + EPSV);

    const float LOG2PI = 1.8378770664093453f;
    for (int d = 0; d < NUM_FEAT; ++d) {
        float m  = mu[k * NUM_FEAT + d];
        float s2 = var[k * NUM_FEAT + d] + EPSV;
        float iv = 1.0f / s2;
        coef[k * 24 + 2 * d]     = -0.5f * iv;   // pairs with x^2
        coef[k * 24 + 2 * d + 1] =  m * iv;      // pairs with x
        c += -0.5f * (LOG2PI + logf(s2)) - 0.5f * m * m * iv;
        mustd[k * 24 + d]      = m;
        mustd[k * 24 + 12 + d] = sqrtf(s2);
    }
    cvec[k] = c;
}

__global__ __launch_bounds__(256) void pd_fused(
    const float* __restrict__ X,
    const float* __restrict__ noise,
    const float* __restrict__ ws,
    float* __restrict__ out,
    int n)
{
    __shared__ float s_mustd[MAX_K * 24];            // 9.6 KB
    __shared__ float s_resp[8][16 * MAX_K];          // 51.2 KB

    const float* coef  = ws + WS_COEF;
    const float* cvec  = ws + WS_CVEC;
    const float* mustd = ws + WS_MUSTD;

    const int tid = threadIdx.x;
    for (int t = tid; t < MAX_K * 24; t += 256) s_mustd[t] = mustd[t];

    const int wave  = tid >> 5;
    const int lane  = tid & 31;
    const int h     = lane >> 4;        // wave half
    const int cl    = lane & 15;        // A-row / B-column within tile
    const int pbase = blockIdx.x * 128 + wave * 16;

    // ---- Phase A: loglik+logw tile via chained V_WMMA_F32_16X16X4_F32 ----
    // A(16x4 f32): lane L holds row M=L%16, K = 4s + 2*(L/16) + {0,1}
    // 24 virtual features [x_d^2, x_d] -> 6 k-steps; K index 4s+2h is even,
    // so this lane always supplies the (x_d^2, x_d) pair for d = 2s+h.
    int prow = pbase + cl; if (prow > n - 1) prow = n - 1;
    const float4* xp4 = (const float4*)(X + (size_t)prow * NUM_FEAT);
    float4 xa = xp4[0], xb = xp4[1], xc = xp4[2];
    float xr[12] = { xa.x, xa.y, xa.z, xa.w,
                     xb.x, xb.y, xb.z, xb.w,
                     xc.x, xc.y, xc.z, xc.w };

    v8f acc[7];
#pragma unroll
    for (int j = 0; j < 7; ++j) {
        float ck = cvec[16 * j + cl];   // per-column constant (broadcast rows)
        v8f a;
#pragma unroll
        for (int r = 0; r < 8; ++r) a[r] = ck;
        acc[j] = a;
    }
#pragma unroll
    for (int s = 0; s < 6; ++s) {
        float xv = h ? xr[2 * s + 1] : xr[2 * s];
        v2f av; av.x = xv * xv; av.y = xv;
#pragma unroll
        for (int j = 0; j < 7; ++j) {
            float2 bq = *(const float2*)(coef + (16 * j + cl) * 24 + 4 * s + 2 * h);
            v2f bv; bv.x = bq.x; bv.y = bq.y;
            acc[j] = __builtin_amdgcn_wmma_f32_16x16x4_f32(
                false, av, false, bv, (short)0, acc[j], false, false);
        }
    }

    // ---- Phase B: softmax over 112 clusters, in-register + half-wave shfl --
    // D entry (VGPR r, lane L) = point pbase + r + 8h, cluster 16j + (L%16)
#pragma unroll
    for (int r = 0; r < 8; ++r) {
        float mx = acc[0][r];
#pragma unroll
        for (int j = 1; j < 7; ++j) mx = fmaxf(mx, acc[j][r]);
#pragma unroll
        for (int msk = 1; msk < 16; msk <<= 1)
            mx = fmaxf(mx, __shfl_xor(mx, msk, 16));
        float sum = 0.0f;
#pragma unroll
        for (int j = 0; j < 7; ++j) {
            float e = __expf(acc[j][r] - mx);
            acc[j][r] = e;
            sum += e;
        }
#pragma unroll
        for (int msk = 1; msk < 16; msk <<= 1)
            sum += __shfl_xor(sum, msk, 16);
        float inv = 1.0f / sum;
#pragma unroll
        for (int j = 0; j < 7; ++j) acc[j][r] *= inv;
    }
#pragma unroll
    for (int j = 0; j < 7; ++j) {
        int k = 16 * j + cl;
        if (k < MAX_K) {
#pragma unroll
            for (int r = 0; r < 8; ++r)
                s_resp[wave][(r + 8 * h) * MAX_K + k] = acc[j][r];
        }
    }
    __syncthreads();   // also publishes s_mustd / s_resp

    // ---- Phase C: stream noise once; 2 lanes per point, interleaved k -----
    const int pl  = lane >> 1;
    const int par = lane & 1;
    const int pt  = pbase + pl;
    float o[12];
#pragma unroll
    for (int d = 0; d < 12; ++d) o[d] = 0.0f;

    if (pt < n) {
        const float4* np =
            (const float4*)(noise + ((size_t)pt * MAX_K + par) * NUM_FEAT);
        for (int k = par; k < MAX_K; k += 2, np += 6) {
            float r = s_resp[wave][pl * MAX_K + k];
            float4 n0 = np[0], n1 = np[1], n2 = np[2];       // global_load_b128
            const float4* mu4 = (const float4*)(s_mustd + k * 24);
            const float4* st4 = (const float4*)(s_mustd + k * 24 + 12);
            float4 m0 = mu4[0], m1 = mu4[1], m2 = mu4[2];
            float4 s0 = st4[0], s1 = st4[1], s2 = st4[2];
            o[0]  += r * fmaf(n0.x, s0.x, m0.x);
            o[1]  += r * fmaf(n0.y, s0.y, m0.y);
            o[2]  += r * fmaf(n0.z, s0.z, m0.z);
            o[3]  += r * fmaf(n0.w, s0.w, m0.w);
            o[4]  += r * fmaf(n1.x, s1.x, m1.x);
            o[5]  += r * fmaf(n1.y, s1.y, m1.y);
            o[6]  += r * fmaf(n1.z, s1.z, m1.z);
            o[7]  += r * fmaf(n1.w, s1.w, m1.w);
            o[8]  += r * fmaf(n2.x, s2.x, m2.x);
            o[9]  += r * fmaf(n2.y, s2.y, m2.y);
            o[10] += r * fmaf(n2.z, s2.z, m2.z);
            o[11] += r * fmaf(n2.w, s2.w, m2.w);
        }
    }
#pragma unroll
    for (int d = 0; d < 12; ++d) o[d] += __shfl_xor(o[d], 1, 32);

    if (!par && pt < n) {
        float4* op = (float4*)(out + (size_t)pt * NUM_FEAT);
        op[0] = make_float4(o[0], o[1], o[2],  o[3]);
        op[1] = make_float4(o[4], o[5], o[6],  o[7]);
        op[2] = make_float4(o[8], o[9], o[10], o[11]);
    }
}

extern "C" void kernel_launch(void* const* d_in, const int* in_sizes, int n_in,
                              void* d_out, int out_size, void* d_ws, size_t ws_size,
                              hipStream_t stream)
{
    const float* X     = (const float*)d_in[0];
    const float* mu    = (const float*)d_in[1];
    const float* var   = (const float*)d_in[2];
    const float* noise = (const float*)d_in[3];
    float* out = (float*)d_out;
    float* ws  = (float*)d_ws;   // needs (112*24 + 112 + 100*24)*4 = ~20.8 KB

    const int n = in_sizes[0] / NUM_FEAT;

    pd_precompute<<<1, 128, 0, stream>>>(mu, var, ws);
    const int blocks = (n + 127) / 128;
    pd_fused<<<blocks, 256, 0, stream>>>(X, noise, ws, out, n);
}